// BatchRankingMSE_Loss_10496900071732
// MI455X (gfx1250) — compile-verified
//
#include <hip/hip_runtime.h>
#include <hip/hip_bf16.h>

typedef __attribute__((ext_vector_type(2))) float v2f;
typedef __attribute__((ext_vector_type(8))) float v8f;

#define RANK_MARGIN 2.0f
#define RANK_BLOCKS 1024
#define MSE_BLOCKS  32

// One wave per 16x16 tile of the (i,j) pair space. Tile (r,c) covers
// i in [r0, r0+16), j in [c0, c0+16); only c >= r tiles are processed.
//
// Pairwise differences on the matrix pipe (V_WMMA_F32_16X16X4_F32):
//   A[m][:] = {1, p_row[m], 0, 0},  B[:][n] = {p_col[n], -1, 0, 0}^T
//   => D[m][n] = p_col[n] - p_row[m]
// One WMMA for pred diffs (Dp), one for label diffs (Dl). Epilogue:
//   x = Dp ^ signbit(Dl);  x = (Dl==0) ? 0 : x;  acc += max(margin - x, 0)
__global__ __launch_bounds__(256) void rank_tiles_kernel(
    const float* __restrict__ preds,
    const float* __restrict__ labels,
    float* __restrict__ partials,
    int n, int nt) {
  const int lane = threadIdx.x & 31;
  const int wib  = threadIdx.x >> 5;                       // wave in block (0..7)
  const int wglobal = blockIdx.x * (blockDim.x >> 5) + wib;
  const int nwaves  = gridDim.x * (blockDim.x >> 5);
  const int ntt = nt * nt;
  const bool lo = lane < 16;
  const int l16 = lane & 15;

  float acc = 0.0f;

  for (int t = wglobal; t < ntt; t += nwaves) {
    const int r = t / nt;
    const int c = t - r * nt;
    if (c < r) continue;                                   // wave-uniform skip
    const int r0 = r << 4;
    const int c0 = c << 4;

    const int ri = r0 + l16;
    const int ci = c0 + l16;
    const float prow = (lo && ri < n) ? preds[ri]  : 0.0f;
    const float pcol = (lo && ci < n) ? preds[ci]  : 0.0f;
    const float lrow = (lo && ri < n) ? labels[ri] : 0.0f;
    const float lcol = (lo && ci < n) ? labels[ci] : 0.0f;

    // f32 A 16x4: lanes 0-15 -> M=lane, {K0,K1}; lanes 16-31 -> {K2,K3}=0
    v2f pA; pA[0] = lo ? 1.0f : 0.0f; pA[1] = prow;
    v2f pB; pB[0] = pcol;             pB[1] = lo ? -1.0f : 0.0f;
    v2f lA; lA[0] = pA[0];            lA[1] = lrow;
    v2f lB; lB[0] = lcol;             lB[1] = pB[1];

    v8f zero = {};
    v8f Dp = __builtin_amdgcn_wmma_f32_16x16x4_f32(
        false, pA, false, pB, (short)0, zero, false, false);
    v8f Dl = __builtin_amdgcn_wmma_f32_16x16x4_f32(
        false, lA, false, lB, (short)0, zero, false, false);

    if (c > r && c0 + 16 <= n) {
      // Fast path: interior off-diagonal tile -> every (i,j) has i < j < n.
#pragma unroll
      for (int v = 0; v < 8; ++v) {
        const float dl = Dl[v];
        unsigned xb = __float_as_uint(Dp[v]) ^
                      (__float_as_uint(dl) & 0x80000000u);
        float x = (dl == 0.0f) ? 0.0f : __uint_as_float(xb);
        acc += fmaxf(RANK_MARGIN - x, 0.0f);
      }
    } else {
      // Diagonal / edge tile: per-element index masking.
      // C layout: lane L, VGPR v -> M = v + 8*(L>=16), N = L%16
      const int Nj  = c0 + l16;                            // global j
      const int hi8 = (lane >> 4) << 3;                    // 0 or 8
#pragma unroll
      for (int v = 0; v < 8; ++v) {
        const int Mi = r0 + v + hi8;                       // global i
        const float dl = Dl[v];
        unsigned xb = __float_as_uint(Dp[v]) ^
                      (__float_as_uint(dl) & 0x80000000u);
        float x = (dl == 0.0f) ? 0.0f : __uint_as_float(xb);
        float tv = fmaxf(RANK_MARGIN - x, 0.0f);
        const bool keep = (Nj > Mi) && (Nj < n);           // i < j < n
        acc += keep ? tv : 0.0f;
      }
    }
  }

  // wave32 reduction (fixed order -> deterministic)
#pragma unroll
  for (int m = 16; m >= 1; m >>= 1)
    acc += __shfl_xor(acc, m, 32);

  __shared__ float sdata[8];
  if (lane == 0) sdata[wib] = acc;
  __syncthreads();
  if (threadIdx.x == 0) {
    float s = 0.0f;
#pragma unroll
    for (int w = 0; w < 8; ++w) s += sdata[w];
    partials[blockIdx.x] = s;                              // write, no atomics
  }
}

__global__ __launch_bounds__(256) void mse_partial_kernel(
    const float* __restrict__ preds,
    const float* __restrict__ labels,
    float* __restrict__ partials, int n) {
  __shared__ float sdata[256];
  float acc = 0.0f;
  for (int i = blockIdx.x * blockDim.x + threadIdx.x; i < n;
       i += gridDim.x * blockDim.x) {
    const float d = preds[i] - labels[i];
    acc += d * d;
  }
  sdata[threadIdx.x] = acc;
  __syncthreads();
  for (int s = 128; s > 0; s >>= 1) {
    if (threadIdx.x < (unsigned)s) sdata[threadIdx.x] += sdata[threadIdx.x + s];
    __syncthreads();
  }
  if (threadIdx.x == 0) partials[blockIdx.x] = sdata[0];
}

__global__ void finalize_kernel(const float* __restrict__ rank_partials, int nrank,
                                const float* __restrict__ mse_partials, int nmse,
                                float* __restrict__ out, int n) {
  if (threadIdx.x == 0 && blockIdx.x == 0) {
    float rs = 0.0f;
    for (int i = 0; i < nrank; ++i) rs += rank_partials[i];
    float ms = 0.0f;
    for (int i = 0; i < nmse; ++i) ms += mse_partials[i];
    const float beta =
        (n == 1) ? 1.0f : 2.0f / ((float)n * (float)(n - 1));
    out[0] = ms / (float)n + beta * rs;
  }
}

extern "C" void kernel_launch(void* const* d_in, const int* in_sizes, int n_in,
                              void* d_out, int out_size, void* d_ws, size_t ws_size,
                              hipStream_t stream) {
  const float* preds  = (const float*)d_in[0];
  const float* labels = (const float*)d_in[1];
  float* out = (float*)d_out;
  const int n  = in_sizes[0];
  const int nt = (n + 15) / 16;

  float* ws     = (float*)d_ws;
  float* rank_p = ws;                 // RANK_BLOCKS floats
  float* mse_p  = ws + RANK_BLOCKS;   // MSE_BLOCKS floats

  rank_tiles_kernel<<<RANK_BLOCKS, 256, 0, stream>>>(preds, labels, rank_p, n, nt);
  mse_partial_kernel<<<MSE_BLOCKS, 256, 0, stream>>>(preds, labels, mse_p, n);
  finalize_kernel<<<1, 32, 0, stream>>>(rank_p, RANK_BLOCKS, mse_p, MSE_BLOCKS, out, n);
}